// Voxelization_19052474925399
// MI455X (gfx1250) — compile-verified
//
#include <hip/hip_runtime.h>
#include <hip/hip_bf16.h>

// ---------------- CDNA5 TDM (Tensor Data Mover) helpers ----------------
typedef unsigned int u32x4 __attribute__((ext_vector_type(4)));
typedef int          i32x8 __attribute__((ext_vector_type(8)));
typedef int          i32x4 __attribute__((ext_vector_type(4)));

// 1-D tensor_load_to_lds: copy `elems` 4-byte elements from global -> LDS.
// Descriptor per CDNA5 ISA ch.8: group0 {count=1, lds_addr, global_addr, type=2},
// group1 {data_size=4B, tensor_dim0=elems, tile_dim0=elems}, groups 2/3 zero (<=2D).
__device__ __forceinline__ void tdm_load_1d(void* lds_dst, const void* gsrc, unsigned elems) {
    // generic "shared" pointer = {SHARED_BASE aperture, lds_offset[31:0]} -> low 32 bits are LDS byte addr
    unsigned lds_addr = (unsigned)(unsigned long long)lds_dst;
    unsigned long long ga = (unsigned long long)gsrc;
    u32x4 g0;
    g0.x = 1u;                                                   // count = 1 (valid descriptor)
    g0.y = lds_addr;                                             // lds_addr [63:32]
    g0.z = (unsigned)(ga & 0xFFFFFFFFull);                       // global_addr lo
    g0.w = (unsigned)((ga >> 32) & 0x01FFFFFFull) | 0x80000000u; // global_addr hi + type=2 [127:126]
    i32x8 g1;
    g1[0] = (int)0x00020000u;                      // data_size = 2 (4 bytes), wg_mask=0
    g1[1] = (int)((elems & 0xFFFFu) << 16);        // tensor_dim0[15:0]  @ bits 63:48
    g1[2] = (int)((elems >> 16) & 0xFFFFu);        // tensor_dim0[31:16] @ bits 79:64
    g1[3] = (int)((elems & 0xFFFFu) << 16);        // tile_dim0          @ bits 127:112
    g1[4] = 0;
    g1[5] = (int)elems;                            // tensor_dim0_stride lo (unused for 1D)
    g1[6] = 0;
    g1[7] = 0;
    i32x4 z4 = {0, 0, 0, 0};
#if defined(__clang_major__) && (__clang_major__ >= 23)
    i32x8 z8 = {0, 0, 0, 0, 0, 0, 0, 0};
    __builtin_amdgcn_tensor_load_to_lds(g0, g1, z4, z4, z8, 0);
#else
    __builtin_amdgcn_tensor_load_to_lds(g0, g1, z4, z4, 0);
#endif
}

// ---------------- problem constants ----------------
#define NOKEY 0xFFFFFFFFu
#define PTS_PER_BLOCK 2048
#define TILE_PTS 1024
#define NWORDS (1 << 18)          // 2^23 voxel keys / 32 bits

// exclusive block scan over 256 values (one per thread), Hillis-Steele in LDS
__device__ __forceinline__ unsigned exscan256(unsigned v, volatile unsigned* s) {
    unsigned tid = threadIdx.x;
    s[tid] = v;
    __syncthreads();
    for (unsigned off = 1; off < 256; off <<= 1) {
        unsigned t = (tid >= off) ? s[tid - off] : 0u;
        __syncthreads();
        s[tid] += t;
        __syncthreads();
    }
    return s[tid] - v;
}

// K0: zero header + bitmap
__global__ __launch_bounds__(256) void k_init(unsigned* W) {
    int n = 16 + NWORDS;
    for (int i = blockIdx.x * blockDim.x + threadIdx.x; i < n; i += gridDim.x * blockDim.x)
        W[i] = 0u;
}

// K1: stream pc via TDM double-buffer, compute keys, mark bitmap, count valid per block
__global__ __launch_bounds__(256) void k_pass1(const float* __restrict__ pc,
                                               const int* __restrict__ batch,
                                               unsigned* __restrict__ bitmap,
                                               unsigned* __restrict__ keys,
                                               unsigned* __restrict__ blockCounts,
                                               int N) {
    __shared__ float sm[2][TILE_PTS * 3];
    __shared__ unsigned red[256];
    const unsigned tid = threadIdx.x;
    const int blockBase = blockIdx.x * PTS_PER_BLOCK;

    int nIssued = 0;
    for (int t = 0; t < 2; ++t)
        if (blockBase + t * TILE_PTS < N) nIssued++;

    if (tid < 32) {  // wave 0 drives the DMA engine (one tensor op per wave)
        for (int t = 0; t < nIssued; ++t) {
            int tbase = blockBase + t * TILE_PTS;
            int npts = min(TILE_PTS, N - tbase);
            tdm_load_1d(&sm[t][0], pc + (size_t)tbase * 3, (unsigned)(npts * 3));
        }
    }

    unsigned cnt = 0;
    for (int t = 0; t < 2; ++t) {
        if (tid < 32) {
            if (t == 0) {
                if (nIssued == 2) __builtin_amdgcn_s_wait_tensorcnt(1);
                else              __builtin_amdgcn_s_wait_tensorcnt(0);
            } else {
                __builtin_amdgcn_s_wait_tensorcnt(0);
            }
        }
        __syncthreads();
        int tbase = blockBase + t * TILE_PTS;
        for (int k = 0; k < 4; ++k) {
            int li = k * 256 + tid;
            int p = tbase + li;
            if (p < N) {
                unsigned key = NOKEY;
                float x = sm[t][li * 3 + 0];
                float y = sm[t][li * 3 + 1];
                float z = sm[t][li * 3 + 2];
                const float eps = 0.0001f;
                bool v = (x > 0.0f + eps) && (x < 51.2f - eps) &&
                         (y > -25.6f + eps) && (y < 25.6f - eps) &&
                         (z > -2.0f + eps) && (z < 4.4f - eps);
                if (v) {
                    int ix = (int)(256.0f * (x - 0.0f) / 51.2f);
                    int iy = (int)(256.0f * (y + 25.6f) / 51.2f);
                    int iz = (int)(32.0f * (z + 2.0f) / 6.4f);
                    ix = min(max(ix, 0), 255);
                    iy = min(max(iy, 0), 255);
                    iz = min(max(iz, 0), 31);
                    int b = batch[p] & 3;
                    key = ((unsigned)b << 21) | ((unsigned)ix << 13) |
                          ((unsigned)iy << 5) | (unsigned)iz;
                    atomicOr(&bitmap[key >> 5], 1u << (key & 31));
                    cnt++;
                }
                keys[p] = key;
            }
        }
        __syncthreads();
    }
    red[tid] = cnt;
    __syncthreads();
    for (int off = 128; off > 0; off >>= 1) {
        if (tid < (unsigned)off) red[tid] += red[tid + off];
        __syncthreads();
    }
    if (tid == 0) blockCounts[blockIdx.x] = red[0];
}

// K2: per-1024-word-block popcount sums
__global__ __launch_bounds__(256) void k_popc(const unsigned* __restrict__ bitmap,
                                              unsigned* __restrict__ wsum) {
    __shared__ unsigned red[256];
    unsigned tid = threadIdx.x;
    unsigned wbase = blockIdx.x * 1024u + tid * 4u;
    unsigned s = 0;
    for (int i = 0; i < 4; ++i) s += (unsigned)__popc(bitmap[wbase + i]);
    red[tid] = s;
    __syncthreads();
    for (int off = 128; off > 0; off >>= 1) {
        if (tid < (unsigned)off) red[tid] += red[tid + off];
        __syncthreads();
    }
    if (tid == 0) wsum[blockIdx.x] = red[0];
}

// K3: tiny serial scans (256 + ~2k elements); U -> hdr[0], M -> hdr[1]
__global__ void k_scan_small(const unsigned* __restrict__ wsum, unsigned* __restrict__ wsumScan,
                             const unsigned* __restrict__ bc, unsigned* __restrict__ bcScan,
                             unsigned* __restrict__ hdr, int nb1) {
    if (threadIdx.x == 0 && blockIdx.x == 0) {
        unsigned acc = 0;
        for (int i = 0; i < 256; ++i) { wsumScan[i] = acc; acc += wsum[i]; }
        hdr[0] = acc;  // U
        acc = 0;
        for (int i = 0; i < nb1; ++i) { bcScan[i] = acc; acc += bc[i]; }
        hdr[1] = acc;  // M
    }
}

// K4: walk set bits in ascending key order -> sorted unique rows + word-rank prefix
__global__ __launch_bounds__(256) void k_emit_unique(const unsigned* __restrict__ bitmap,
                                                     const unsigned* __restrict__ wsumScan,
                                                     unsigned* __restrict__ wordPrefix,
                                                     int* __restrict__ out) {
    __shared__ unsigned sbuf[256];
    unsigned tid = threadIdx.x;
    unsigned wbase = blockIdx.x * 1024u + tid * 4u;
    unsigned pcs = 0;
    unsigned wv[4];
    for (int i = 0; i < 4; ++i) { wv[i] = bitmap[wbase + i]; pcs += (unsigned)__popc(wv[i]); }
    unsigned off = exscan256(pcs, sbuf);
    unsigned cur = wsumScan[blockIdx.x] + off;
    for (int i = 0; i < 4; ++i) {
        unsigned w = wbase + i;
        wordPrefix[w] = cur;
        unsigned m = wv[i];
        while (m) {
            int j = __builtin_ctz(m);
            m &= m - 1u;
            unsigned key = (w << 5) | (unsigned)j;
            int* row = out + (size_t)cur * 4;
            row[0] = (int)(key >> 21);
            row[1] = (int)((key >> 13) & 255u);
            row[2] = (int)((key >> 5) & 255u);
            row[3] = (int)(key & 31u);
            cur++;
        }
    }
}

// K5: inverse indices for the compacted (order-preserving) filtered points
__global__ __launch_bounds__(256) void k_emit_inv(const unsigned* __restrict__ keys,
                                                  const unsigned* __restrict__ bitmap,
                                                  const unsigned* __restrict__ wordPrefix,
                                                  const unsigned* __restrict__ bcScan,
                                                  const unsigned* __restrict__ hdr,
                                                  int* __restrict__ out, int N) {
    __shared__ unsigned sbuf[256];
    unsigned tid = threadIdx.x;
    int base = blockIdx.x * PTS_PER_BLOCK + (int)tid * 8;
    unsigned k[8];
    unsigned c = 0;
    for (int i = 0; i < 8; ++i) {
        int p = base + i;
        k[i] = (p < N) ? keys[p] : NOKEY;
        c += (k[i] != NOKEY) ? 1u : 0u;
    }
    unsigned off = exscan256(c, sbuf);
    unsigned pos = bcScan[blockIdx.x] + off;
    unsigned U = hdr[0];
    int* inv = out + (size_t)U * 4;
    for (int i = 0; i < 8; ++i) {
        if (k[i] != NOKEY) {
            unsigned key = k[i];
            unsigned w = key >> 5;
            unsigned rank = wordPrefix[w] +
                            (unsigned)__popc(bitmap[w] & ((1u << (key & 31u)) - 1u));
            inv[pos++] = (int)rank;
        }
    }
}

extern "C" void kernel_launch(void* const* d_in, const int* in_sizes, int n_in,
                              void* d_out, int out_size, void* d_ws, size_t ws_size,
                              hipStream_t stream) {
    const float* pc = (const float*)d_in[0];
    const int* batch = (const int*)d_in[1];
    int N = in_sizes[0] / 3;
    int* out = (int*)d_out;

    unsigned* W = (unsigned*)d_ws;
    int nb1 = (N + PTS_PER_BLOCK - 1) / PTS_PER_BLOCK;
    unsigned* hdr = W;                       // 16
    unsigned* bitmap = hdr + 16;             // NWORDS
    unsigned* wordPrefix = bitmap + NWORDS;  // NWORDS
    unsigned* wsum = wordPrefix + NWORDS;    // 256
    unsigned* wsumScan = wsum + 256;         // 256
    unsigned* bc = wsumScan + 256;           // nb1
    unsigned* bcScan = bc + nb1;             // nb1
    unsigned* keys = bcScan + nb1;           // N

    k_init<<<1024, 256, 0, stream>>>(hdr);
    k_pass1<<<nb1, 256, 0, stream>>>(pc, batch, bitmap, keys, bc, N);
    k_popc<<<256, 256, 0, stream>>>(bitmap, wsum);
    k_scan_small<<<1, 32, 0, stream>>>(wsum, wsumScan, bc, bcScan, hdr, nb1);
    k_emit_unique<<<256, 256, 0, stream>>>(bitmap, wsumScan, wordPrefix, out);
    k_emit_inv<<<nb1, 256, 0, stream>>>(keys, bitmap, wordPrefix, bcScan, hdr, out, N);
}